// ReconModelOrig_17549236371726
// MI455X (gfx1250) — compile-verified
//
#include <hip/hip_runtime.h>
#include <hip/hip_bf16.h>

// ---------------- problem constants (from reference) ----------------
#define BB   256
#define NV   35709
#define NF   70789
#define DEG  8
#define NTOT (3 * NV)        // 107127
#define CSTRIDE 257
#define MBLK 4               // M-tiles (of 16 batch rows) per wave

typedef __attribute__((ext_vector_type(2))) float v2f;
typedef __attribute__((ext_vector_type(8))) float v8f;

// ---------------------------------------------------------------
// Kernel 0: mean3 reduction + per-batch rotation matrix and SH gains
// One block of 256 threads. Thread t handles batch t after reduction.
// ---------------------------------------------------------------
__global__ __launch_bounds__(256) void prep_kernel(
    const float* __restrict__ coeff,      // [B,257]
    const float* __restrict__ meanshape,  // [3*NV]
    float* __restrict__ rot,              // [B,9]  rot[k*3+c] for n@R
    float* __restrict__ g,                // [B,9,3]
    float* __restrict__ mean3)            // [3]
{
    __shared__ float sm[3 * 256];
    const int tid = threadIdx.x;

    // --- mean of meanshape per component ---
    float s0 = 0.f, s1 = 0.f, s2 = 0.f;
    for (int k = tid; k < NV; k += 256) {
        s0 += meanshape[3 * k + 0];
        s1 += meanshape[3 * k + 1];
        s2 += meanshape[3 * k + 2];
    }
    sm[tid] = s0; sm[256 + tid] = s1; sm[512 + tid] = s2;
    __syncthreads();
    for (int off = 128; off > 0; off >>= 1) {
        if (tid < off) {
            sm[tid]       += sm[tid + off];
            sm[256 + tid] += sm[256 + tid + off];
            sm[512 + tid] += sm[512 + tid + off];
        }
        __syncthreads();
    }
    if (tid == 0) {
        mean3[0] = sm[0]   * (1.0f / NV);
        mean3[1] = sm[256] * (1.0f / NV);
        mean3[2] = sm[512] * (1.0f / NV);
    }

    // --- per-batch rotation + gains ---
    const int b = tid;  // B == 256 == blockDim
    const float* c = coeff + b * CSTRIDE;
    float ax = c[224], ay = c[225], az = c[226];
    float sx = __sinf(ax), cx = __cosf(ax);
    float sy = __sinf(ay), cy = __cosf(ay);
    float sz = __sinf(az), cz = __cosf(az);
    // M = Rz*Ry*Rx ; rotation = M^T ; (n @ rotation)[c] = sum_k n[k]*M[c][k]
    float M[3][3];
    M[0][0] = cz * cy;  M[0][1] = -sz * cx + cz * sy * sx;  M[0][2] = sz * sx + cz * sy * cx;
    M[1][0] = sz * cy;  M[1][1] =  cz * cx + sz * sy * sx;  M[1][2] = -cz * sx + sz * sy * cx;
    M[2][0] = -sy;      M[2][1] =  cy * sx;                 M[2][2] = cy * cx;
    float* r = rot + b * 9;
    #pragma unroll
    for (int k = 0; k < 3; ++k)
        #pragma unroll
        for (int cc = 0; cc < 3; ++cc)
            r[k * 3 + cc] = M[cc][k];

    // g[b][k][c] = gamma[c*9+k] + (k==0 ? 0.8 : 0)
    const float* gam = c + 227;
    float* go = g + b * 27;
    #pragma unroll
    for (int k = 0; k < 9; ++k)
        #pragma unroll
        for (int cc = 0; cc < 3; ++cc)
            go[k * 3 + cc] = gam[cc * 9 + k] + (k == 0 ? 0.8f : 0.0f);
}

// ---------------------------------------------------------------
// Kernel 1: fp32 WMMA GEMM, register-blocked 4x over M.
//   out[m, i] = sum_k coeff[m, cof1+k]*base1[i,k]  (+ base2 term)
//              + meanvec[i] (- mean3[i%3] if mean3 != null)
// One wave32 per (64-row M strip) x (16-col N tile); K stepped by 4 via
// V_WMMA_F32_16X16X4_F32. Each B float2 load feeds MBLK=4 WMMAs.
// Per ISA 7.12.2 (32-bit A 16x4 / B 4x16): lanes 0-15 hold K={0,1},
// lanes 16-31 hold K={2,3}; C/D: lane=N, VGPR r -> M=r (+8 for upper lanes).
// ---------------------------------------------------------------
__global__ __launch_bounds__(256) void gemm_wmma_kernel(
    const float* __restrict__ coeff,   // [B,257]
    const float* __restrict__ base1, int ld1, int cof1,
    const float* __restrict__ base2, int ld2, int cof2,   // base2 may be null
    const float* __restrict__ meanvec, // [NTOT]
    const float* __restrict__ mean3,   // [3] or null
    float* __restrict__ out)           // [B, NTOT]
{
    const int lane   = threadIdx.x & 31;
    const int warp   = threadIdx.x >> 5;
    const int wid    = blockIdx.x * 8 + warp;
    const int tilesN = (NTOT + 15) >> 4;
    const int mgrp   = wid & (MBLK - 1);     // 4 strips of 64 rows cover B=256
    const int tile_n = wid >> 2;
    if (tile_n >= tilesN) return;            // uniform per wave

    const int sub  = lane & 15;
    const int half = lane >> 4;              // 0: K={0,1}, 1: K={2,3}
    const int nb   = tile_n * 16;
    const int i    = min(nb + sub, NTOT - 1);   // clamped base row

    const float* arow[MBLK];
    #pragma unroll
    for (int t = 0; t < MBLK; ++t)
        arow[t] = coeff + (mgrp * (MBLK * 16) + t * 16 + sub) * CSTRIDE;

    v8f acc[MBLK] = {};

    {
        const float* brow = base1 + (size_t)i * ld1;
        for (int k = 0; k < ld1; k += 4) {
            const int kk = k + half * 2;
            v2f bv = *(const v2f*)(brow + kk);
            #pragma unroll
            for (int t = 0; t < MBLK; ++t) {
                v2f a; a.x = arow[t][cof1 + kk]; a.y = arow[t][cof1 + kk + 1];
                acc[t] = __builtin_amdgcn_wmma_f32_16x16x4_f32(
                    false, a, false, bv, (short)0, acc[t], false, false);
            }
        }
    }
    if (base2) {
        const float* brow = base2 + (size_t)i * ld2;
        for (int k = 0; k < ld2; k += 4) {
            const int kk = k + half * 2;
            v2f bv = *(const v2f*)(brow + kk);
            #pragma unroll
            for (int t = 0; t < MBLK; ++t) {
                v2f a; a.x = arow[t][cof2 + kk]; a.y = arow[t][cof2 + kk + 1];
                acc[t] = __builtin_amdgcn_wmma_f32_16x16x4_f32(
                    false, a, false, bv, (short)0, acc[t], false, false);
            }
        }
    }

    float addv = meanvec[i];
    if (mean3) addv -= mean3[i % 3];

    const int n = nb + sub;
    if (n < NTOT) {
        #pragma unroll
        for (int t = 0; t < MBLK; ++t) {
            #pragma unroll
            for (int r = 0; r < 8; ++r) {
                const int mm = mgrp * (MBLK * 16) + t * 16 + r + half * 8;
                out[(size_t)mm * NTOT + n] = acc[t][r] + addv;
            }
        }
    }
}

// ---------------------------------------------------------------
// Kernel 2: per-(batch, face) normals; row NF is the zero pad row.
// ---------------------------------------------------------------
__global__ __launch_bounds__(256) void facenorm_kernel(
    const float* __restrict__ fs,   // [B, 3*NV] centered face_shape
    const int*   __restrict__ tri,  // [NF,3]
    float* __restrict__ fn)         // [B, NF+1, 3]
{
    const long t = (long)blockIdx.x * blockDim.x + threadIdx.x;
    const long total = (long)BB * (NF + 1);
    if (t >= total) return;
    const int b = (int)(t / (NF + 1));
    const int f = (int)(t - (long)b * (NF + 1));
    float* o = fn + ((size_t)b * (NF + 1) + f) * 3;
    if (f == NF) { o[0] = 0.f; o[1] = 0.f; o[2] = 0.f; return; }

    const int v0 = tri[f * 3 + 0], v1 = tri[f * 3 + 1], v2 = tri[f * 3 + 2];
    const float* s = fs + (size_t)b * NTOT;
    const float p0x = s[3 * v0], p0y = s[3 * v0 + 1], p0z = s[3 * v0 + 2];
    const float p1x = s[3 * v1], p1y = s[3 * v1 + 1], p1z = s[3 * v1 + 2];
    const float p2x = s[3 * v2], p2y = s[3 * v2 + 1], p2z = s[3 * v2 + 2];
    const float e1x = p0x - p1x, e1y = p0y - p1y, e1z = p0z - p1z;
    const float e2x = p1x - p2x, e2y = p1y - p2y, e2z = p1z - p2z;
    float nx = e1y * e2z - e1z * e2y;
    float ny = e1z * e2x - e1x * e2z;
    float nz = e1x * e2y - e1y * e2x;
    const float nrm = sqrtf(nx * nx + ny * ny + nz * nz);
    const float inv = 1.0f / fmaxf(nrm, 1e-12f);
    o[0] = nx * inv; o[1] = ny * inv; o[2] = nz * inv;
}

// ---------------------------------------------------------------
// Kernel 3: per-(batch, vertex) vertex normal (8-way gather) ->
// normalize -> rotate -> SH lighting -> color -> output transform.
// ---------------------------------------------------------------
__global__ __launch_bounds__(256) void shade_kernel(
    const float* __restrict__ fn,        // [B, NF+1, 3]
    const int*   __restrict__ point_buf, // [NV, DEG]
    const float* __restrict__ rot,       // [B,9]
    const float* __restrict__ g,         // [B,9,3]
    const float* __restrict__ tex,       // [B, 3*NV]
    float* __restrict__ out)             // [B, NV, 3]
{
    const long t = (long)blockIdx.x * blockDim.x + threadIdx.x;
    if (t >= (long)BB * NV) return;
    const int b = (int)(t / NV);
    const int v = (int)(t - (long)b * NV);

    const float* fnb = fn + (size_t)b * (NF + 1) * 3;
    float sx = 0.f, sy = 0.f, sz = 0.f;
    #pragma unroll
    for (int d = 0; d < DEG; ++d) {
        const int idx = point_buf[v * DEG + d];
        const float* p = fnb + (size_t)idx * 3;
        sx += p[0]; sy += p[1]; sz += p[2];
    }
    const float nrm = sqrtf(sx * sx + sy * sy + sz * sz);
    const float inv = 1.0f / fmaxf(nrm, 1e-12f);
    sx *= inv; sy *= inv; sz *= inv;

    const float* r = rot + b * 9;
    const float nx = sx * r[0] + sy * r[3] + sz * r[6];
    const float ny = sx * r[1] + sy * r[4] + sz * r[7];
    const float nz = sx * r[2] + sy * r[5] + sz * r[8];

    // SH basis constants: a0c0=sqrt(pi)/2, a1c1=sqrt(pi), a2c2=2.4270335
    const float A0 = 0.8862269254527580f;
    const float A1 = 1.7724538509055159f;
    const float A2 = 2.4270335120834243f;
    float Y[9];
    Y[0] = A0;
    Y[1] = -A1 * ny;
    Y[2] =  A1 * nz;
    Y[3] = -A1 * nx;
    Y[4] =  A2 * nx * ny;
    Y[5] = -A2 * ny * nz;
    Y[6] =  (0.5f * A2 / 1.7320508075688772f) * (3.0f * nz * nz - 1.0f);
    Y[7] = -A2 * nx * nz;
    Y[8] =  0.5f * A2 * (nx * nx - ny * ny);

    const float* gb = g + b * 27;
    float l0 = 0.f, l1 = 0.f, l2 = 0.f;
    #pragma unroll
    for (int k = 0; k < 9; ++k) {
        l0 += Y[k] * gb[k * 3 + 0];
        l1 += Y[k] * gb[k * 3 + 1];
        l2 += Y[k] * gb[k * 3 + 2];
    }

    const float* tx = tex + (size_t)b * NTOT + 3 * (size_t)v;
    float* o = out + (size_t)b * NV * 3 + 3 * (size_t)v;
    const float k128 = 1.0f / 128.0f;
    o[0] = (tx[0] * l0 - 127.5f) * k128;
    o[1] = (tx[1] * l1 - 127.5f) * k128;
    o[2] = (tx[2] * l2 - 127.5f) * k128;
}

// ---------------------------------------------------------------
// launch
// ---------------------------------------------------------------
extern "C" void kernel_launch(void* const* d_in, const int* in_sizes, int n_in,
                              void* d_out, int out_size, void* d_ws, size_t ws_size,
                              hipStream_t stream) {
    (void)in_sizes; (void)n_in; (void)out_size; (void)ws_size;
    const float* coeff     = (const float*)d_in[0];
    const float* idBase    = (const float*)d_in[1];
    const float* exBase    = (const float*)d_in[2];
    const float* texBase   = (const float*)d_in[3];
    const float* meanshape = (const float*)d_in[4];
    const float* meantex   = (const float*)d_in[5];
    const int*   tri       = (const int*)d_in[6];
    const int*   point_buf = (const int*)d_in[7];
    float* out = (float*)d_out;

    // workspace carve-out (256B aligned slices)
    char* ws = (char*)d_ws;
    size_t off = 0;
    auto carve = [&](size_t bytes) -> char* {
        char* p = ws + off;
        off += (bytes + 255) & ~(size_t)255;
        return p;
    };
    float* rot   = (float*)carve((size_t)BB * 9 * sizeof(float));
    float* gg    = (float*)carve((size_t)BB * 27 * sizeof(float));
    float* mean3 = (float*)carve(16);
    float* fs    = (float*)carve((size_t)BB * NTOT * sizeof(float));   // ~110 MB
    float* ft    = (float*)carve((size_t)BB * NTOT * sizeof(float));   // ~110 MB
    float* fn    = (float*)carve((size_t)BB * (NF + 1) * 3 * sizeof(float)); // ~217 MB

    // 0: prep (mean3 + rotation + gains)
    prep_kernel<<<1, 256, 0, stream>>>(coeff, meanshape, rot, gg, mean3);

    // 1: WMMA GEMMs (one wave = 64 batch rows x 16 base rows)
    const int tilesN = (NTOT + 15) / 16;
    const int blocksG = (tilesN * MBLK + 7) / 8;   // 8 waves per block
    // face_shape = idBase@id + exBase@ex + meanshape - mean3
    gemm_wmma_kernel<<<blocksG, 256, 0, stream>>>(
        coeff, idBase, 80, 0, exBase, 64, 80, meanshape, mean3, fs);
    // face_texture = texBase@tex + meantex
    gemm_wmma_kernel<<<blocksG, 256, 0, stream>>>(
        coeff, texBase, 80, 144, nullptr, 0, 0, meantex, nullptr, ft);

    // 2: face normals (+ zero pad row)
    {
        const long total = (long)BB * (NF + 1);
        const int blocks = (int)((total + 255) / 256);
        facenorm_kernel<<<blocks, 256, 0, stream>>>(fs, tri, fn);
    }

    // 3: vertex normals + rotate + SH lighting + color
    {
        const long total = (long)BB * NV;
        const int blocks = (int)((total + 255) / 256);
        shade_kernel<<<blocks, 256, 0, stream>>>(fn, point_buf, rot, gg, ft, out);
    }
}